// GAT_54760833024266
// MI455X (gfx1250) — compile-verified
//
#include <hip/hip_runtime.h>

typedef __attribute__((ext_vector_type(2))) float v2f;
typedef __attribute__((ext_vector_type(8))) float v8f;

#define HID 64

// ---------------------------------------------------------------------------
// Kernel 1: s_i = x @ w_i, s_j = x @ w_j via V_WMMA_F32_16X16X4_F32.
// One wave per 16-node tile. A = x[16 nodes x 4 K], B = [4 x 16] with
// column 0 = w_i, column 1 = w_j, rest zero. 16 WMMAs sweep K=64.
// B built branchlessly: unconditional b64 loads + 0/1 select masks, so no
// EXEC-mask churn between WMMAs.
// ---------------------------------------------------------------------------
__global__ void k_sij_wmma(const float* __restrict__ x,
                           const float* __restrict__ w_i,
                           const float* __restrict__ w_j,
                           float* __restrict__ s_i,
                           float* __restrict__ s_j,
                           int n, int ntiles) {
  int wave = blockIdx.x * (blockDim.x >> 5) + (threadIdx.x >> 5);
  if (wave >= ntiles) return;                 // wave-uniform exit
  int lane = threadIdx.x & 31;
  int h    = lane >> 4;                       // wave half
  int m    = lane & 15;                       // A-matrix row M
  int ncol = lane & 15;                       // B/D column N
  int row  = wave * 16 + m;
  if (row >= n) row = n - 1;                  // clamp (N is a multiple of 16 here)
  const float* xrow = x + (size_t)row * HID;

  float sel_i = (ncol == 0) ? 1.0f : 0.0f;    // B column selectors (branchless)
  float sel_j = (ncol == 1) ? 1.0f : 0.0f;

  v8f c = {};                                 // 16x16 f32 accumulator
#pragma unroll
  for (int kt = 0; kt < HID / 4; ++kt) {
    int kk = kt * 4 + 2 * h;                  // lanes 0-15: K0,K1; lanes 16-31: K2,K3
    v2f a   = *(const v2f*)(xrow + kk);       // global_load_b64
    v2f wi2 = *(const v2f*)(w_i + kk);        // unconditional, L0-resident
    v2f wj2 = *(const v2f*)(w_j + kk);
    v2f b;
    b.x = sel_i * wi2.x + sel_j * wj2.x;
    b.y = sel_i * wi2.y + sel_j * wj2.y;
    c = __builtin_amdgcn_wmma_f32_16x16x4_f32(false, a, false, b,
                                              (short)0, c, false, false);
  }
  // D layout: VGPR v -> M = v + 8*h, N = lane&15. Only columns 0,1 are useful.
  if (ncol < 2) {
    float* dstv = (ncol == 0) ? s_i : s_j;
#pragma unroll
    for (int v = 0; v < 8; ++v) {
      int r = wave * 16 + v + 8 * h;
      if (r < n) dstv[r] = c[v];
    }
  }
}

// ---------------------------------------------------------------------------
// Kernel 2: in-degree histogram (real edges only; self-loops handled later).
// ---------------------------------------------------------------------------
__global__ void k_hist(const int* __restrict__ dst, int* __restrict__ deg, int E) {
  int e = blockIdx.x * blockDim.x + threadIdx.x;
  if (e < E) atomicAdd(&deg[dst[e]], 1);
}

// ---------------------------------------------------------------------------
// Kernel 3: single-block exclusive scan -> row_ptr, and cursor = row_ptr copy.
// ---------------------------------------------------------------------------
__global__ void k_scan(const int* __restrict__ deg, int* __restrict__ row_ptr,
                       int* __restrict__ cursor, int n) {
  __shared__ int sm[1024];
  __shared__ int s_carry;
  int t = threadIdx.x;
  if (t == 0) s_carry = 0;
  __syncthreads();
  for (int base = 0; base < n; base += 1024) {
    int v = (base + t < n) ? deg[base + t] : 0;
    sm[t] = v;
    __syncthreads();
    for (int off = 1; off < 1024; off <<= 1) {
      int tmp = (t >= off) ? sm[t - off] : 0;
      __syncthreads();
      sm[t] += tmp;
      __syncthreads();
    }
    int carry = s_carry;
    int excl  = carry + sm[t] - v;
    if (base + t < n) { row_ptr[base + t] = excl; cursor[base + t] = excl; }
    __syncthreads();
    if (t == 1023) s_carry = carry + sm[1023];
    __syncthreads();
  }
  if (t == 0) row_ptr[n] = s_carry;
}

// ---------------------------------------------------------------------------
// Kernel 4: scatter edge sources into CSR column array.
// ---------------------------------------------------------------------------
__global__ void k_scatter(const int* __restrict__ src, const int* __restrict__ dst,
                          int* __restrict__ cursor, int* __restrict__ col, int E) {
  int e = blockIdx.x * blockDim.x + threadIdx.x;
  if (e < E) {
    int d   = dst[e];
    int pos = atomicAdd(&cursor[d], 1);
    col[pos] = src[e];
  }
}

// ---------------------------------------------------------------------------
// Kernel 5: fused segment-softmax + SpMM + ReLU. One wave per destination.
// Lane f owns features f and f+32. Zero FP atomics; one store per output.
// Self-loop handled implicitly (not stored in CSR).
// ---------------------------------------------------------------------------
__device__ __forceinline__ float lrelu(float e) {
  return e >= 0.0f ? e : 0.01f * e;
}

__global__ void k_gat(const float* __restrict__ x,
                      const float* __restrict__ s_i, const float* __restrict__ s_j,
                      const int* __restrict__ row_ptr, const int* __restrict__ col,
                      float* __restrict__ out, int n) {
  int wave = blockIdx.x * (blockDim.x >> 5) + (threadIdx.x >> 5);
  if (wave >= n) return;                      // wave-uniform
  int lane  = threadIdx.x & 31;
  int d     = wave;
  int start = row_ptr[d];
  int end   = row_ptr[d + 1];
  float si  = s_i[d];

  float e_self = lrelu(si + s_j[d]);
  // Pass 1: segment max (lane-parallel over neighbors, then wave reduce).
  float m = e_self;
  for (int k = start + lane; k < end; k += 32)
    m = fmaxf(m, lrelu(si + s_j[col[k]]));
#pragma unroll
  for (int off = 16; off > 0; off >>= 1)
    m = fmaxf(m, __shfl_xor(m, off, 32));

  // Pass 2: exp-sum + weighted feature accumulation in one sweep.
  float w     = __expf(e_self - m);
  float denom = w;
  const float* xr = x + (size_t)d * HID;
  float acc0 = w * xr[lane];
  float acc1 = w * xr[lane + 32];
  for (int k = start; k < end; ++k) {
    int s    = col[k];
    float wk = __expf(lrelu(si + s_j[s]) - m);
    denom += wk;
    const float* xs = x + (size_t)s * HID;
    acc0 += wk * xs[lane];
    acc1 += wk * xs[lane + 32];
  }
  float inv = 1.0f / denom;
  out[(size_t)d * HID + lane]      = fmaxf(acc0 * inv, 0.0f);
  out[(size_t)d * HID + lane + 32] = fmaxf(acc1 * inv, 0.0f);
}

// ---------------------------------------------------------------------------
extern "C" void kernel_launch(void* const* d_in, const int* in_sizes, int n_in,
                              void* d_out, int out_size, void* d_ws, size_t ws_size,
                              hipStream_t stream) {
  const float* x  = (const float*)d_in[0];
  const int*   ei = (const int*)d_in[1];
  const float* wi = (const float*)d_in[2];
  const float* wj = (const float*)d_in[3];
  float* out = (float*)d_out;

  int N = in_sizes[0] / HID;
  int E = in_sizes[1] / 2;
  const int* src = ei;       // row 0 = j (source)
  const int* dst = ei + E;   // row 1 = i (target)

  // Workspace carve-up (~8.3 MB total).
  char* ws = (char*)d_ws;
  auto alignup = [](size_t v) { return (v + 255) & ~(size_t)255; };
  size_t o = 0;
  float* s_i = (float*)(ws + o); o = alignup(o + (size_t)N * 4);
  float* s_j = (float*)(ws + o); o = alignup(o + (size_t)N * 4);
  int*   deg = (int*)(ws + o);   o = alignup(o + (size_t)N * 4);
  int*   rp  = (int*)(ws + o);   o = alignup(o + (size_t)(N + 1) * 4);
  int*   cur = (int*)(ws + o);   o = alignup(o + (size_t)N * 4);
  int*   col = (int*)(ws + o);   o = alignup(o + (size_t)E * 4);
  (void)ws_size; (void)n_in; (void)out_size;

  hipMemsetAsync(deg, 0, (size_t)N * 4, stream);

  int ntiles = (N + 15) / 16;
  int wpb = 8;  // 8 waves per 256-thread block
  k_sij_wmma<<<(ntiles + wpb - 1) / wpb, 256, 0, stream>>>(x, wi, wj, s_i, s_j, N, ntiles);
  k_hist   <<<(E + 255) / 256, 256, 0, stream>>>(dst, deg, E);
  k_scan   <<<1, 1024, 0, stream>>>(deg, rp, cur, N);
  k_scatter<<<(E + 255) / 256, 256, 0, stream>>>(src, dst, cur, col, E);
  k_gat    <<<(N + wpb - 1) / wpb, 256, 0, stream>>>(x, s_i, s_j, rp, col, out, N);
}